// CustomCTCLoss_69690139344987
// MI455X (gfx1250) — compile-verified
//
#include <hip/hip_runtime.h>
#include <math.h>

// Problem constants from the reference
#define T_DIM 512
#define N_DIM 256
#define C_DIM 256
#define S_DIM 48
#define L_DIM (2 * S_DIM + 1)   // 97 extended-lattice states
#define NEGV  (-1e30f)

// ---- CDNA5 async global->LDS path (guarded; fallback = synchronous copy) ----
#if defined(__has_builtin)
#  if __has_builtin(__builtin_amdgcn_global_load_async_to_lds_b64)
#    define HAVE_ASYNC_LDS 1
#  endif
#  if __has_builtin(__builtin_amdgcn_s_wait_asynccnt)
#    define HAVE_WAIT_ASYNC_BUILTIN 1
#  endif
#endif
#ifndef HAVE_ASYNC_LDS
#  define HAVE_ASYNC_LDS 0
#endif
#ifndef HAVE_WAIT_ASYNC_BUILTIN
#  define HAVE_WAIT_ASYNC_BUILTIN 0
#endif

// b64 variant moves a gcc-vector int2 per lane; global src is AS1, LDS dst is AS3.
typedef int v2i __attribute__((vector_size(2 * sizeof(int))));

// Address-space casts via integer round-trip (flat LDS addr low 32 bits == LDS offset)
#define AS1V2(p) ((__attribute__((address_space(1))) v2i*)(unsigned long long)(p))
#define AS3V2(p) ((__attribute__((address_space(3))) v2i*)(unsigned int)(unsigned long long)(p))

__device__ __forceinline__ void wait_async_zero() {
#if HAVE_ASYNC_LDS
#  if HAVE_WAIT_ASYNC_BUILTIN
    __builtin_amdgcn_s_wait_asynccnt(0);
#  else
    asm volatile("s_wait_asynccnt 0" ::: "memory");
#  endif
#endif
}

// Stage one 256-float emission row into LDS: 128 threads x 8 bytes = 1 KB.
__device__ __forceinline__ void stage_row(const float* __restrict__ g,
                                          float* __restrict__ l, int tid) {
#if HAVE_ASYNC_LDS
    __builtin_amdgcn_global_load_async_to_lds_b64(AS1V2(g + 2 * tid), AS3V2(l + 2 * tid),
                                                  /*imm offset*/0, /*cpol*/0);
#else
    float2 v = *(const float2*)(g + 2 * tid);
    *(float2*)(l + 2 * tid) = v;
#endif
}

// log(exp(a)+exp(b)) without NaNs for finite -1e30 sentinels
__device__ __forceinline__ float lae(float a, float b) {
    float m = fmaxf(a, b);
    float d = fminf(a, b) - m;          // <= 0
    return m + log1pf(__expf(d));
}

__global__ __launch_bounds__(128)
void ctc_alpha_kernel(const float* __restrict__ logp,   // (T, N, C) log-probs
                      const int*   __restrict__ labels, // (N, S)
                      const int*   __restrict__ in_len, // (N,)
                      const int*   __restrict__ tgt_len,// (N,)
                      float*       __restrict__ loss_per_n) // (N,)
{
    const int n   = blockIdx.x;
    const int tid = threadIdx.x;

    __shared__ float alpha[2 + 128];                  // [0..1] = -inf pad, state l at [2+l]
    __shared__ __align__(16) float lp_row[2][C_DIM];  // double-buffered emission rows
    __shared__ int   lab_s[S_DIM];

    if (tid < S_DIM) lab_s[tid] = labels[n * S_DIM + tid];
    if (tid < 2)     alpha[tid] = NEGV;
    alpha[2 + tid] = NEGV;

    // Stage emission row t=0
    stage_row(logp + (size_t)n * C_DIM, lp_row[0], tid);
    wait_async_zero();
    __syncthreads();   // lab_s, alpha init, lp_row[0] all visible

    // Extended label and diagonal-skip mask for this state
    int ext = 0, skip = 0;
    if (tid < L_DIM && (tid & 1)) {
        int k = tid >> 1;                 // odd l -> label index (l-1)/2
        ext  = lab_s[k];
        skip = (tid >= 3) && (lab_s[k] != lab_s[k - 1]);
    }

    const int ilen = in_len[n];
    const int tlen = tgt_len[n];

    // alpha0
    if (tid == 0) alpha[2] = lp_row[0][0];     // blank
    if (tid == 1) alpha[3] = lp_row[0][ext];   // first label
    __syncthreads();

    // Prefetch row t=1 while nothing depends on it yet
    stage_row(logp + ((size_t)N_DIM + n) * C_DIM, lp_row[1], tid);

    for (int t = 1; t < T_DIM; ++t) {
        // read neighbors of previous alpha before it is overwritten
        float a  = alpha[2 + tid];
        float s1 = alpha[1 + tid];
        float s2 = alpha[tid];

        wait_async_zero();   // emission row t has landed in LDS (this wave's DMA)
        __syncthreads();     // all waves' DMA portions + alpha reads complete

        float nv = a;        // freeze past input_length (and for lanes >= L)
        if (tid < L_DIM && t < ilen) {
            float comb = lae(a, s1);
            if (skip) comb = lae(comb, s2);
            nv = comb + lp_row[t & 1][ext];
        }

        // Issue DMA for row t+1 into the buffer last read at step t-1 (safe past barrier)
        if (t + 1 < T_DIM)
            stage_row(logp + ((size_t)(t + 1) * N_DIM + n) * C_DIM,
                      lp_row[(t + 1) & 1], tid);

        if (tid < L_DIM) alpha[2 + tid] = nv;
        __syncthreads();
    }

    if (tid == 0) {
        float la   = alpha[2 + 2 * tlen - 1];
        float lb   = alpha[2 + 2 * tlen];
        float loss = -lae(la, lb);
        if (loss > 1e29f) loss = 0.0f;             // zero_infinity
        loss_per_n[n] = loss / (float)tlen;
    }
}

// Deterministic tree reduction of the 256 per-batch losses -> mean
__global__ __launch_bounds__(N_DIM)
void ctc_reduce_kernel(const float* __restrict__ per_n, float* __restrict__ out)
{
    __shared__ float buf[N_DIM];
    const int tid = threadIdx.x;
    buf[tid] = per_n[tid];
    __syncthreads();
    for (int s = N_DIM / 2; s > 0; s >>= 1) {
        if (tid < s) buf[tid] += buf[tid + s];
        __syncthreads();
    }
    if (tid == 0) {
        float loss = buf[0] / (float)N_DIM;
        if (isnan(loss) || isinf(loss)) loss = 0.0f;  // sanitize()
        out[0] = loss;
    }
}

extern "C" void kernel_launch(void* const* d_in, const int* in_sizes, int n_in,
                              void* d_out, int out_size, void* d_ws, size_t ws_size,
                              hipStream_t stream) {
    const float* logits     = (const float*)d_in[0];  // (T, N, C) fp32
    const int*   labels     = (const int*)d_in[1];    // (N, S) int32
    const int*   pred_sizes = (const int*)d_in[2];    // (N,)
    const int*   targ_sizes = (const int*)d_in[3];    // (N,)
    float*       out        = (float*)d_out;          // scalar loss
    float*       per_n      = (float*)d_ws;           // N floats of scratch

    ctc_alpha_kernel<<<N_DIM, 128, 0, stream>>>(logits, labels, pred_sizes, targ_sizes, per_n);
    ctc_reduce_kernel<<<1, N_DIM, 0, stream>>>(per_n, out);
}